// PadVariable_41231686042251
// MI455X (gfx1250) — compile-verified
//
#include <hip/hip_runtime.h>

// Problem constants from the reference:
//   x: (N, T, F) fp32, lens: (N,), pad: (2, N), out: (N, Tp, F) fp32
#define N_CONST 64
#define T_CONST 4096
#define F_CONST 64

// Native clang vector type: accepted by nontemporal builtins and lowers to
// global_load_b128 / global_store_b128.
typedef __attribute__((ext_vector_type(4))) float v4f;

// Block = 256 threads (8 wave32s). 16 lanes x v4f cover one F=64 row (256 B),
// so one block handles 16 consecutive t-rows of a single sequence n and
// writes 4 KB of contiguous output -> perfectly coalesced B128 traffic.
__global__ __launch_bounds__(256) void pad_reflect_gather_kernel(
    const float* __restrict__ x,     // (N, T, F)
    const int*   __restrict__ lens,  // (N,)
    const int*   __restrict__ pad,   // (2, N) flattened
    float*       __restrict__ out,   // (N, Tp, F)
    int Tp)
{
    const int n = blockIdx.y;                       // uniform -> SGPR
    // Uniform metadata: compiler emits scalar loads, broadcast to the wave.
    const int L    = lens[n];
    const int p0   = pad[n];                        // pad[0][n]
    const int p1   = pad[N_CONST + n];              // pad[1][n]
    const int vlen = L + p0 + p1;                   // valid length

    const int tid = threadIdx.x;
    const int row = tid >> 4;                       // 0..15: which t-row
    const int col = (tid & 15) << 2;                // float column 0,4,...,60
    const int t   = blockIdx.x * 16 + row;
    if (t >= Tp) return;

    v4f v = (v4f){0.0f, 0.0f, 0.0f, 0.0f};
    if (t < vlen) {
        // Reflection index: rel<0 -> -rel ; rel<L -> rel ; else 2L-rel-2
        const int rel = t - p0;
        int src = (rel < 0) ? -rel : ((rel < L) ? rel : (2 * L - rel - 2));
        // clip to [0, T-1]
        src = src < 0 ? 0 : src;
        src = src > (T_CONST - 1) ? (T_CONST - 1) : src;

        const v4f* xp = reinterpret_cast<const v4f*>(
            x + ((size_t)n * T_CONST + (size_t)src) * F_CONST + col);
        v = *xp;                                    // global_load_b128 (RT:
                                                    // edge rows re-read; input
                                                    // fits in 192MB L2)
    }

    // Output is write-once, never re-read by this kernel -> non-temporal
    // B128 store to avoid displacing the (reused) input in cache.
    v4f* op = reinterpret_cast<v4f*>(
        out + ((size_t)n * (size_t)Tp + (size_t)t) * F_CONST + col);
    __builtin_nontemporal_store(v, op);             // global_store_b128 (NT)
}

extern "C" void kernel_launch(void* const* d_in, const int* in_sizes, int n_in,
                              void* d_out, int out_size, void* d_ws, size_t ws_size,
                              hipStream_t stream) {
    (void)in_sizes; (void)n_in; (void)d_ws; (void)ws_size;

    const float* x    = (const float*)d_in[0];
    const int*   lens = (const int*)d_in[1];
    const int*   pad  = (const int*)d_in[2];
    // d_in[3] is Tp as a 1-element device array; we need it on the host for
    // grid sizing, so recover it from out_size = N * Tp * F.
    const int Tp = out_size / (N_CONST * F_CONST);

    dim3 block(256, 1, 1);
    dim3 grid((Tp + 15) / 16, N_CONST, 1);
    pad_reflect_gather_kernel<<<grid, block, 0, stream>>>(
        x, lens, pad, (float*)d_out, Tp);
}